// Net_28312424415416
// MI455X (gfx1250) — compile-verified
//
#include <hip/hip_runtime.h>

typedef __attribute__((ext_vector_type(2))) float v2f;
typedef __attribute__((ext_vector_type(8))) float v8f;

#define N_NODES 100000
#define N_EDGES 1600000

// ---------------- degree / normalization ----------------

__global__ void fill_kernel(float* p, int n, float v) {
    int i = blockIdx.x * blockDim.x + threadIdx.x;
    if (i < n) p[i] = v;
}

__global__ void deg_kernel(const long long* __restrict__ dst, float* __restrict__ deg, int e_cnt) {
    int e = blockIdx.x * blockDim.x + threadIdx.x;
    if (e < e_cnt) unsafeAtomicAdd(&deg[(int)dst[e]], 1.0f);
}

__global__ void rsqrt_kernel(float* p, int n) {
    int i = blockIdx.x * blockDim.x + threadIdx.x;
    if (i < n) p[i] = rsqrtf(p[i]);
}

// ---------------- fp32 WMMA GEMM: H[nrows,DOUT] = X[nrows,K] @ W[K,DOUT] ----------------
// One wave computes a 16 x DOUT strip (NCT = ceil(DOUT/16) column tiles), reusing the
// A fragment across all column tiles so X is streamed exactly once.
// V_WMMA_F32_16X16X4_F32 layout (ISA 7.12.2):
//   A 16x4:  lanes 0-15 -> (row=lane, K=0 in v0, K=1 in v1); lanes 16-31 -> K=2,3
//   B 4x16:  lanes 0-15 -> (col=lane, K=0 in v0, K=1 in v1); lanes 16-31 -> K=2,3
//   C/D 16x16: v8f, element r of lane l -> row = r + 8*(l>=16), col = l&15
template <int K, int DOUT, int NCT>
__global__ void gemm_wmma_f32(const float* __restrict__ X, const float* __restrict__ W,
                              float* __restrict__ H, int nrows) {
    int wave = (blockIdx.x * blockDim.x + threadIdx.x) >> 5;
    int lane = threadIdx.x & 31;
    int row0 = wave * 16;
    if (row0 >= nrows) return;  // wave-uniform: EXEC stays all-ones for WMMA

    const int half = lane >> 4;   // 0: K pair {0,1}; 1: K pair {2,3}
    const int lr   = lane & 15;   // row (for A) / col (for B, C/D)

    v8f acc[NCT] = {};

    const float* xrow = X + (size_t)(row0 + lr) * K;

#pragma unroll 4
    for (int k = 0; k < K; k += 4) {
        v2f a;
        a.x = xrow[k + half * 2 + 0];
        a.y = xrow[k + half * 2 + 1];
#pragma unroll
        for (int ct = 0; ct < NCT; ++ct) {
            int col = ct * 16 + lr;
            v2f b;
            b.x = (col < DOUT) ? W[(k + half * 2 + 0) * DOUT + col] : 0.0f;
            b.y = (col < DOUT) ? W[(k + half * 2 + 1) * DOUT + col] : 0.0f;
            acc[ct] = __builtin_amdgcn_wmma_f32_16x16x4_f32(
                false, a, false, b, (short)0, acc[ct], false, false);
        }
    }

#pragma unroll
    for (int ct = 0; ct < NCT; ++ct) {
        int col = ct * 16 + lr;
        if (col < DOUT) {
#pragma unroll
            for (int r = 0; r < 8; ++r) {
                H[(size_t)(row0 + half * 8 + r) * DOUT + col] = acc[ct][r];
            }
        }
    }
}

// ---------------- edge scatter: agg[dst] += h[src] * dinv[src]*dinv[dst] ----------------
// One thread per (edge, 8-feature chunk): two float4 gathers + 8 fp32 global atomics.
template <int D>
__global__ void scatter_edges(const long long* __restrict__ src, const long long* __restrict__ dst,
                              const float* __restrict__ h, const float* __restrict__ dinv,
                              float* __restrict__ agg, int e_cnt) {
    constexpr int CH = D / 8;
    int t = blockIdx.x * blockDim.x + threadIdx.x;
    int e = t / CH;
    int c = t % CH;
    if (e >= e_cnt) return;
    int s = (int)src[e];
    int d = (int)dst[e];
    float coef = dinv[s] * dinv[d];
    const float4* hp = (const float4*)(h + (size_t)s * D + c * 8);
    float4 v0 = hp[0];
    float4 v1 = hp[1];
    float* ap = agg + (size_t)d * D + c * 8;
    unsafeAtomicAdd(ap + 0, v0.x * coef);
    unsafeAtomicAdd(ap + 1, v0.y * coef);
    unsafeAtomicAdd(ap + 2, v0.z * coef);
    unsafeAtomicAdd(ap + 3, v0.w * coef);
    unsafeAtomicAdd(ap + 4, v1.x * coef);
    unsafeAtomicAdd(ap + 5, v1.y * coef);
    unsafeAtomicAdd(ap + 6, v1.z * coef);
    unsafeAtomicAdd(ap + 7, v1.w * coef);
}

// ---------------- self-loop + bias (+ ReLU), in-place on `out` ----------------
__global__ void self_bias_act(const float* __restrict__ h, const float* __restrict__ bias,
                              const float* __restrict__ dinv, float* __restrict__ out,
                              int n, int d, int do_relu) {
    int idx = blockIdx.x * blockDim.x + threadIdx.x;
    if (idx >= n * d) return;
    int i = idx / d;
    int f = idx % d;
    float di = dinv[i];
    float v = out[idx] + h[idx] * di * di + bias[f];
    out[idx] = do_relu ? fmaxf(v, 0.0f) : v;
}

// ---------------- launch ----------------

extern "C" void kernel_launch(void* const* d_in, const int* in_sizes, int n_in,
                              void* d_out, int out_size, void* d_ws, size_t ws_size,
                              hipStream_t stream) {
    (void)in_sizes; (void)n_in; (void)out_size; (void)ws_size;

    const int N = N_NODES, E = N_EDGES;

    const float* x  = (const float*)d_in[0];
    const long long* ei = (const long long*)d_in[1];  // int64 [2, E]
    const float* W1 = (const float*)d_in[2];
    const float* b1 = (const float*)d_in[3];
    const float* W2 = (const float*)d_in[4];
    const float* b2 = (const float*)d_in[5];
    const float* W3 = (const float*)d_in[6];
    const float* b3 = (const float*)d_in[7];
    float* out = (float*)d_out;

    const long long* src = ei;
    const long long* dst = ei + E;

    // workspace: h-buffer, two aggregation buffers, d_inv
    float* buf_h = (float*)d_ws;           // N*64
    float* buf_a = buf_h + (size_t)N * 64; // N*64
    float* buf_b = buf_a + (size_t)N * 64; // N*64
    float* dinv  = buf_b + (size_t)N * 64; // N

    const int T = 256;
    const int strips = N / 16;                 // 6250 (exact)
    const int gemm_blocks = (strips + 7) / 8;  // 8 waves/block

    // degrees (with self-loop) -> d_inv
    fill_kernel<<<(N + T - 1) / T, T, 0, stream>>>(dinv, N, 1.0f);
    deg_kernel<<<(E + T - 1) / T, T, 0, stream>>>(dst, dinv, E);
    rsqrt_kernel<<<(N + T - 1) / T, T, 0, stream>>>(dinv, N);

    // ---- layer 1: 128 -> 64, relu ----
    gemm_wmma_f32<128, 64, 4><<<gemm_blocks, T, 0, stream>>>(x, W1, buf_h, N);
    hipMemsetAsync(buf_a, 0, (size_t)N * 64 * sizeof(float), stream);
    scatter_edges<64><<<(E * 8 + T - 1) / T, T, 0, stream>>>(src, dst, buf_h, dinv, buf_a, E);
    self_bias_act<<<(N * 64 + T - 1) / T, T, 0, stream>>>(buf_h, b1, dinv, buf_a, N, 64, 1);

    // ---- layer 2: 64 -> 64, relu ----
    gemm_wmma_f32<64, 64, 4><<<gemm_blocks, T, 0, stream>>>(buf_a, W2, buf_h, N);
    hipMemsetAsync(buf_b, 0, (size_t)N * 64 * sizeof(float), stream);
    scatter_edges<64><<<(E * 8 + T - 1) / T, T, 0, stream>>>(src, dst, buf_h, dinv, buf_b, E);
    self_bias_act<<<(N * 64 + T - 1) / T, T, 0, stream>>>(buf_h, b2, dinv, buf_b, N, 64, 1);

    // ---- layer 3: 64 -> 40, no relu, straight into d_out ----
    gemm_wmma_f32<64, 40, 3><<<gemm_blocks, T, 0, stream>>>(buf_b, W3, buf_h, N);
    hipMemsetAsync(out, 0, (size_t)N * 40 * sizeof(float), stream);
    scatter_edges<40><<<(E * 5 + T - 1) / T, T, 0, stream>>>(src, dst, buf_h, dinv, out, E);
    self_bias_act<<<(N * 40 + T - 1) / T, T, 0, stream>>>(buf_h, b3, dinv, out, N, 40, 0);
}